// GeneralizedAdvantageEstimation_987842478408
// MI455X (gfx1250) — compile-verified
//
#include <hip/hip_runtime.h>
#include <math.h>

// ---------------- problem constants (from reference) ----------------
#define T_STEPS 2048
#define N_ENVS  8192            // power of two: 2^13
#define N_SHIFT 13
#define GAMMA_F   0.99f
#define LAMBDA_F  0.95f
#define EPS_F     1e-8f

#define CHUNKS  64              // T split into 64 chunks
#define CLEN    32              // 2048 / 64

typedef float v2f __attribute__((ext_vector_type(2)));
typedef float v8f __attribute__((ext_vector_type(8)));

// ---------------- pass 0: zero the double accumulators ----------------
__global__ void gae_init_acc(double* __restrict__ acc) {
    if (threadIdx.x == 0) { acc[0] = 0.0; acc[1] = 0.0; }
}

// ---------------- pass 1: per-chunk backward partials ----------------
// thread (j, n): a_partial = backward scan of chunk j col n with carry 0,
//                P = product of coefficients over the chunk.
__global__ void gae_pass1(const float* __restrict__ r,
                          const float* __restrict__ v,
                          const int*   __restrict__ dn,
                          float* __restrict__ pA,
                          float* __restrict__ pP) {
    const int tid = blockIdx.x * blockDim.x + threadIdx.x;   // CHUNKS*N threads
    const int n = tid & (N_ENVS - 1);
    const int j = tid >> N_SHIFT;
    const int tstart = j * CLEN;

    float a = 0.0f;
    float P = 1.0f;
    float v_next = v[(size_t)(tstart + CLEN) * N_ENVS + n];  // V_{t+1} at chunk top
    #pragma unroll 4
    for (int t = tstart + CLEN - 1; t >= tstart; --t) {
        const size_t row = (size_t)t * N_ENVS + n;
        const float rt = r[row];
        const float vt = v[row];
        const float nt = 1.0f - (float)dn[row];
        const float delta = rt + GAMMA_F * v_next * nt - vt;
        const float c = (GAMMA_F * LAMBDA_F) * nt;
        a = delta + c * a;
        P *= c;
        v_next = vt;
        // gfx1250 global_prefetch_b8: pull earlier rows toward WGP$
        int pt = t - 8; if (pt < tstart) pt = tstart;
        const size_t prow = (size_t)pt * N_ENVS + n;
        __builtin_prefetch(&r[prow], 0, 0);
        __builtin_prefetch(&v[prow], 0, 0);
        __builtin_prefetch(&dn[prow], 0, 0);
    }
    pA[tid] = a;
    pP[tid] = P;
}

// ---------------- pass 2: scan chunk summaries (carry per column) ----------------
__global__ void gae_pass2(const float* __restrict__ pA,
                          const float* __restrict__ pP,
                          float* __restrict__ cin) {
    const int n = blockIdx.x * blockDim.x + threadIdx.x;     // N threads
    float carry = 0.0f;
    for (int j = CHUNKS - 1; j >= 0; --j) {
        const int idx = j * N_ENVS + n;
        cin[idx] = carry;                 // carry entering chunk j (advantage at its end)
        carry = pA[idx] + pP[idx] * carry;
    }
}

// ---------------- pass 3: replay chunks with correct carry, emit adv + returns ----------------
__global__ void gae_pass3(const float* __restrict__ r,
                          const float* __restrict__ v,
                          const int*   __restrict__ dn,
                          const float* __restrict__ cin,
                          float* __restrict__ out) {          // [0,T*N): raw adv, [T*N,2T*N): returns
    const int tid = blockIdx.x * blockDim.x + threadIdx.x;    // CHUNKS*N threads
    const int n = tid & (N_ENVS - 1);
    const int j = tid >> N_SHIFT;
    const int tstart = j * CLEN;

    float a = cin[tid];
    float v_next = v[(size_t)(tstart + CLEN) * N_ENVS + n];
    #pragma unroll 4
    for (int t = tstart + CLEN - 1; t >= tstart; --t) {
        const size_t row = (size_t)t * N_ENVS + n;
        const float rt = r[row];
        const float vt = v[row];
        const float nt = 1.0f - (float)dn[row];
        const float delta = rt + GAMMA_F * v_next * nt - vt;
        const float c = (GAMMA_F * LAMBDA_F) * nt;
        a = delta + c * a;
        out[row] = a;                                          // raw advantage
        out[(size_t)T_STEPS * N_ENVS + row] = a + vt;          // return
        v_next = vt;
        int pt = t - 8; if (pt < tstart) pt = tstart;
        const size_t prow = (size_t)pt * N_ENVS + n;
        __builtin_prefetch(&r[prow], 0, 0);
        __builtin_prefetch(&v[prow], 0, 0);
        __builtin_prefetch(&dn[prow], 0, 0);
    }
}

// ---------------- pass 4: WMMA f32 reduction of sum / sumsq ----------------
// D = A x Ones + C : every column of D holds fp32 row-sums of the 16x4 A tile,
// so each v_wmma_f32_16x16x4_f32 tree-adds 64 fp32 elements into the fp32 C tile.
// Total = (sum over all 256 accumulator entries) / 16 (16 identical columns).
__global__ void gae_reduce_wmma(const float* __restrict__ adv,
                                double* __restrict__ acc) {
    const int total = gridDim.x * blockDim.x;                  // 262144
    const int tid = blockIdx.x * blockDim.x + threadIdx.x;
    const v2f* a2 = (const v2f*)adv;
    const int PAIRS = (T_STEPS * N_ENVS) / 2;                  // 8388608
    const int iters = PAIRS / (1024 * 256);                    // 32, exact -> EXEC all-ones

    v2f ones; ones.x = 1.0f; ones.y = 1.0f;
    v8f accS = {}; v8f accQ = {};

    int i = tid;
    for (int k = 0; k < iters; ++k, i += total) {
        v2f x = a2[i];
        accS = __builtin_amdgcn_wmma_f32_16x16x4_f32(
            false, x,  false, ones, (short)0, accS, false, false);
        v2f xs; xs.x = x.x * x.x; xs.y = x.y * x.y;
        accQ = __builtin_amdgcn_wmma_f32_16x16x4_f32(
            false, xs, false, ones, (short)0, accQ, false, false);
    }

    float s = accS[0] + accS[1] + accS[2] + accS[3] + accS[4] + accS[5] + accS[6] + accS[7];
    float q = accQ[0] + accQ[1] + accQ[2] + accQ[3] + accQ[4] + accQ[5] + accQ[6] + accQ[7];
    // wave32 cross-lane reduce
    for (int off = 16; off > 0; off >>= 1) {
        s += __shfl_down(s, off, 32);
        q += __shfl_down(q, off, 32);
    }
    if ((threadIdx.x & 31) == 0) {
        atomicAdd(&acc[0], (double)(s * (1.0f / 16.0f)));
        atomicAdd(&acc[1], (double)(q * (1.0f / 16.0f)));
    }
}

// ---------------- pass 5: normalize advantages in place ----------------
__global__ void gae_normalize(float* __restrict__ adv,
                              const double* __restrict__ acc) {
    const double cnt = (double)T_STEPS * (double)N_ENVS;
    const double S = acc[0];
    const double Q = acc[1];
    const double mean_d = S / cnt;
    double var = (Q - S * S / cnt) / (cnt - 1.0);
    if (var < 0.0) var = 0.0;
    const float m = (float)mean_d;
    const float inv = 1.0f / ((float)sqrt(var) + EPS_F);

    const int total = gridDim.x * blockDim.x;                  // 262144
    const int tid = blockIdx.x * blockDim.x + threadIdx.x;
    float4* a4 = (float4*)adv;
    const int V = (T_STEPS * N_ENVS) / 4;                      // 4194304
    const int iters = V / (1024 * 256);                        // 16, exact

    int i = tid;
    for (int k = 0; k < iters; ++k, i += total) {
        float4 x = a4[i];
        x.x = (x.x - m) * inv;
        x.y = (x.y - m) * inv;
        x.z = (x.z - m) * inv;
        x.w = (x.w - m) * inv;
        a4[i] = x;
    }
}

// ---------------- launch ----------------
extern "C" void kernel_launch(void* const* d_in, const int* in_sizes, int n_in,
                              void* d_out, int out_size, void* d_ws, size_t ws_size,
                              hipStream_t stream) {
    (void)in_sizes; (void)n_in; (void)out_size; (void)ws_size;

    const float* rewards = (const float*)d_in[0];   // [T, N]
    const float* values  = (const float*)d_in[1];   // [T+1, N]
    const int*   dones   = (const int*)d_in[2];     // [T, N]
    float* out = (float*)d_out;                     // [2*T*N]

    // workspace layout: [0,256): double acc[2]; then pA, pP, cin (CHUNKS*N floats each)
    double* acc = (double*)d_ws;
    float* pA  = (float*)((char*)d_ws + 256);
    float* pP  = pA + CHUNKS * N_ENVS;
    float* cin = pP + CHUNKS * N_ENVS;

    const int BT = 256;
    gae_init_acc<<<1, 64, 0, stream>>>(acc);
    gae_pass1<<<(CHUNKS * N_ENVS) / BT, BT, 0, stream>>>(rewards, values, dones, pA, pP);
    gae_pass2<<<N_ENVS / BT, BT, 0, stream>>>(pA, pP, cin);
    gae_pass3<<<(CHUNKS * N_ENVS) / BT, BT, 0, stream>>>(rewards, values, dones, cin, out);
    gae_reduce_wmma<<<1024, BT, 0, stream>>>(out, acc);
    gae_normalize<<<1024, BT, 0, stream>>>(out, acc);
}